// FrameTransformerDecoder_67113158967447
// MI455X (gfx1250) — compile-verified
//
#include <hip/hip_runtime.h>
#include <math.h>

// ---------------------------------------------------------------------------
// FrameTransformerDecoder for MI455X (gfx1250, wave32).
// All GEMMs use V_WMMA_F32_16X16X4_F32 (complex GEMM = 4 real WMMA chains).
// Workload is HBM-bound (~1GB traffic vs ~260 GFLOP), so f32 WMMA keeps full
// precision at zero roofline cost.
// ---------------------------------------------------------------------------

typedef __attribute__((ext_vector_type(2))) float v2f;
typedef __attribute__((ext_vector_type(8))) float v8f;

#define B_   2
#define CI_  4
#define CO_  8
#define F_   1024
#define W_   256
#define H_   8
#define D_   128            // F_/H_
#define E_   2048
#define EPS_ 1e-6f
#define INV_SCALE_ 0.03125f // 1/sqrt(F_) = 1/32

// ---------------------------------------------------------------------------
// Core: one wave computes a 16x16 complex C tile:  C = sum_k A[m,k]*B[k,n].
// Strides are in complex elements. Fragment layouts per CDNA5 ISA 7.12.2:
//  A 16x4 f32 : lanes 0-15 rows m, VGPR0=K0/VGPR1=K1 ; lanes16-31 K2/K3.
//  B  4x16 f32: VGPR0 = rows k0,k0+2 across lane halves; VGPR1 = k0+1,k0+3.
//  C 16x16 f32: VGPR j -> rows j (lanes 0-15) and j+8 (lanes 16-31), col=lane%16.
// ---------------------------------------------------------------------------
__device__ __forceinline__ void cgemm_tile(
    const float2* __restrict__ A, int la_m, int la_k,
    const float2* __restrict__ B, int lb_k, int lb_n,
    int K, v8f& sr, v8f& si)
{
  v8f rr = {}, ii = {}, ri = {}, ir = {};
  const int lane = threadIdx.x & 31;
  const int half = lane >> 4;
  const int l16  = lane & 15;
  const float2* __restrict__ Arow = A + (size_t)l16 * la_m;
  const float2* __restrict__ Bcol = B + (size_t)l16 * lb_n;
  for (int k0 = 0; k0 < K; k0 += 4) {
    const int ka = k0 + (half << 1);
    float2 a0 = Arow[(size_t)(ka + 0) * la_k];
    float2 a1 = Arow[(size_t)(ka + 1) * la_k];
    float2 b0 = Bcol[(size_t)(ka + 0) * lb_k];
    float2 b1 = Bcol[(size_t)(ka + 1) * lb_k];
    if (k0 + 4 < K) {
      __builtin_prefetch(&Arow[(size_t)(ka + 4) * la_k], 0, 1);
      __builtin_prefetch(&Bcol[(size_t)(ka + 4) * lb_k], 0, 1);
    }
    v2f ar = {a0.x, a1.x}, ai = {a0.y, a1.y};
    v2f br = {b0.x, b1.x}, bi = {b0.y, b1.y};
    rr = __builtin_amdgcn_wmma_f32_16x16x4_f32(false, ar, false, br, (short)0, rr, false, false);
    ii = __builtin_amdgcn_wmma_f32_16x16x4_f32(false, ai, false, bi, (short)0, ii, false, false);
    ri = __builtin_amdgcn_wmma_f32_16x16x4_f32(false, ar, false, bi, (short)0, ri, false, false);
    ir = __builtin_amdgcn_wmma_f32_16x16x4_f32(false, ai, false, br, (short)0, ir, false, false);
  }
  sr = rr - ii;   // real part
  si = ri + ir;   // imag part
}

// ---------------------------------------------------------------------------
// 1) embed: h[b,o,f,w] = sum_ci ew[o,ci]*x[b,ci,f,w] + eb[o]; also copy x into
//    the first CI_ channels of the concat output.
// ---------------------------------------------------------------------------
__global__ void embed_kernel(const float2* __restrict__ x, const float2* __restrict__ ew,
                             const float2* __restrict__ eb, float2* __restrict__ h,
                             float2* __restrict__ outc)
{
  size_t idx = (size_t)blockIdx.x * blockDim.x + threadIdx.x;  // over B*F*W
  if (idx >= (size_t)B_ * F_ * W_) return;
  int b = (int)(idx / ((size_t)F_ * W_));
  size_t fw = idx % ((size_t)F_ * W_);
  float2 xi[CI_];
#pragma unroll
  for (int ci = 0; ci < CI_; ++ci) {
    xi[ci] = x[((size_t)b * CI_ + ci) * F_ * W_ + fw];
    outc[((size_t)b * (CI_ + CO_) + ci) * F_ * W_ + fw] = xi[ci];
  }
#pragma unroll
  for (int o = 0; o < CO_; ++o) {
    float2 acc = eb[o];
#pragma unroll
    for (int ci = 0; ci < CI_; ++ci) {
      float2 w = ew[o * CI_ + ci];
      acc.x += xi[ci].x * w.x - xi[ci].y * w.y;
      acc.y += xi[ci].x * w.y + xi[ci].y * w.x;
    }
    h[((size_t)b * CO_ + o) * F_ * W_ + fw] = acc;
  }
}

// ---------------------------------------------------------------------------
// 2) Complex multichannel LayerNorm over features (axis F, stride W in memory).
//    Block = 256 threads as (32 w-lanes x 8 f-rows); coalesced along w.
// ---------------------------------------------------------------------------
__global__ void cln_kernel(const float2* __restrict__ xin, const float2* __restrict__ nw,
                           const float2* __restrict__ nb, float2* __restrict__ yout)
{
  __shared__ float sre[8][32], sim[8][32], ssq[8][32];
  __shared__ float mure[32], muim[32], sinv[32];
  const int tw = threadIdx.x & 31;
  const int tf = threadIdx.x >> 5;         // 0..7
  const int w0 = blockIdx.x * 32;
  const int bc = blockIdx.y;               // b*CO_ + c
  const int c  = bc % CO_;
  const float2* __restrict__ Xp = xin + (size_t)bc * F_ * W_;
  float ar = 0.f, ai = 0.f, a2 = 0.f;
  for (int f = tf; f < F_; f += 8) {
    float2 v = Xp[(size_t)f * W_ + w0 + tw];
    ar += v.x; ai += v.y; a2 += v.x * v.x + v.y * v.y;
  }
  sre[tf][tw] = ar; sim[tf][tw] = ai; ssq[tf][tw] = a2;
  __syncthreads();
  if (tf == 0) {
    float r = 0.f, i = 0.f, q = 0.f;
#pragma unroll
    for (int j = 0; j < 8; ++j) { r += sre[j][tw]; i += sim[j][tw]; q += ssq[j][tw]; }
    float mr = r / (float)F_, mi = i / (float)F_;
    float var = q / (float)F_ - (mr * mr + mi * mi);
    mure[tw] = mr; muim[tw] = mi; sinv[tw] = rsqrtf(var + EPS_);
  }
  __syncthreads();
  const float mr = mure[tw], mi = muim[tw], iv = sinv[tw];
  float2* __restrict__ Yp = yout + (size_t)bc * F_ * W_;
  for (int f = tf; f < F_; f += 8) {
    float2 v = Xp[(size_t)f * W_ + w0 + tw];
    float xr = (v.x - mr) * iv, xi2 = (v.y - mi) * iv;
    float2 wv = nw[(size_t)c * F_ + f];
    float2 bv = nb[(size_t)c * F_ + f];
    Yp[(size_t)f * W_ + w0 + tw] =
        make_float2(xr * wv.x - xi2 * wv.y + bv.x, xr * wv.y + xi2 * wv.x + bv.y);
  }
}

// ---------------------------------------------------------------------------
// 3) Complex 3x3 SAME conv, direct with LDS input tile (16x16 out per block).
// ---------------------------------------------------------------------------
__global__ void cconv3_kernel(const float2* __restrict__ xin, const float2* __restrict__ wt,
                              const float2* __restrict__ bias, float2* __restrict__ yout)
{
  __shared__ float2 tile[18][18];
  __shared__ float2 wsh[CO_][9];
  const int bz = blockIdx.z;               // b*CO_ + o
  const int b = bz / CO_, o = bz % CO_;
  const int f0 = blockIdx.x * 16;
  const int w0 = blockIdx.y * 16;
  const int t = threadIdx.x;               // 256
  const int tf = t >> 4, tw = t & 15;
  if (t < CO_ * 9) wsh[t / 9][t % 9] = wt[(size_t)o * CO_ * 9 + t];
  float accr = 0.f, acci = 0.f;
  for (int ci = 0; ci < CO_; ++ci) {
    __syncthreads();
    for (int idx = t; idx < 18 * 18; idx += 256) {
      int lf = idx / 18, lw = idx % 18;
      int gf = f0 + lf - 1, gw = w0 + lw - 1;
      float2 v = make_float2(0.f, 0.f);
      if (gf >= 0 && gf < F_ && gw >= 0 && gw < W_)
        v = xin[((size_t)b * CO_ + ci) * F_ * W_ + (size_t)gf * W_ + gw];
      tile[lf][lw] = v;
    }
    __syncthreads();
#pragma unroll
    for (int df = 0; df < 3; ++df)
#pragma unroll
      for (int dw = 0; dw < 3; ++dw) {
        float2 a = tile[tf + df][tw + dw];
        float2 k = wsh[ci][df * 3 + dw];
        accr += a.x * k.x - a.y * k.y;
        acci += a.x * k.y + a.y * k.x;
      }
  }
  float2 bb = bias[o];
  yout[((size_t)b * CO_ + o) * F_ * W_ + (size_t)(f0 + tf) * W_ + (w0 + tw)] =
      make_float2(accr + bb.x, acci + bb.y);
}

// ---------------------------------------------------------------------------
// 4) Attention scores: s[q,k] = (1/32) * sum_d Q[q,d]*K[k,d]  (complex, WMMA).
//    Q/K stored head-split in (F,W) planes: element (w,dd) at (h*D+dd)*W + w.
//    Optionally write |s| and angle(s) to qkm/qkp outputs (attention 1).
// ---------------------------------------------------------------------------
__global__ void scores_kernel(const float2* __restrict__ q, const float2* __restrict__ k,
                              float2* __restrict__ s, float* __restrict__ qkm,
                              float* __restrict__ qkp, int write_mp)
{
  const int bz = blockIdx.y;                 // (b*CO_+c)*H_ + h
  const int tile = blockIdx.x;               // (W/16)^2 tiles
  const int tm = (tile / (W_ / 16)) * 16;    // query tile
  const int tn = (tile % (W_ / 16)) * 16;    // key tile
  const int bc = bz / H_, hh = bz % H_;
  const float2* qbase = q + ((size_t)bc * F_ + (size_t)hh * D_) * W_;
  const float2* kbase = k + ((size_t)bc * F_ + (size_t)hh * D_) * W_;
  v8f sr, si;
  // A: Q[m=w, k=dd] stride(m)=1, stride(k)=W ; B: K^T[k=dd, n=kpos] stride(k)=W, stride(n)=1
  cgemm_tile(qbase + tm, 1, W_, kbase + tn, W_, 1, D_, sr, si);
  const int lane = threadIdx.x & 31, half = lane >> 4, l16 = lane & 15;
  const size_t srow = (size_t)bz * W_ * W_;
#pragma unroll
  for (int j = 0; j < 8; ++j) {
    int m = tm + j + 8 * half, n = tn + l16;
    float re = sr[j] * INV_SCALE_, im = si[j] * INV_SCALE_;
    size_t off = srow + (size_t)m * W_ + n;
    s[off] = make_float2(re, im);
    if (write_mp) {
      qkm[off] = sqrtf(re * re + im * im);
      qkp[off] = atan2f(im, re);
    }
  }
}

// ---------------------------------------------------------------------------
// 5) w = softmax(|s|, axis=k) * s/|s|   (in place; one 256-wide row per block)
// ---------------------------------------------------------------------------
__global__ void softmax_phase_kernel(float2* __restrict__ s)
{
  __shared__ float red[256];
  const int row = blockIdx.x;                // B*CO*H*W rows
  const int t = threadIdx.x;                 // 256 == W_
  float2 v = s[(size_t)row * W_ + t];
  const float mag = sqrtf(v.x * v.x + v.y * v.y);
  red[t] = mag; __syncthreads();
  for (int st = 128; st > 0; st >>= 1) { if (t < st) red[t] = fmaxf(red[t], red[t + st]); __syncthreads(); }
  const float mx = red[0]; __syncthreads();
  const float e = __expf(mag - mx);
  red[t] = e; __syncthreads();
  for (int st = 128; st > 0; st >>= 1) { if (t < st) red[t] += red[t + st]; __syncthreads(); }
  const float p = e / red[0];
  float2 o;
  if (mag > 1e-30f) { float iv = p / mag; o = make_float2(v.x * iv, v.y * iv); }
  else              { o = make_float2(p, 0.f); }
  s[(size_t)row * W_ + t] = o;
}

// ---------------------------------------------------------------------------
// 6) a[q,dd] = sum_k w[q,k]*V[k,dd]  (complex WMMA); write merged-head layout.
// ---------------------------------------------------------------------------
__global__ void av_kernel(const float2* __restrict__ s, const float2* __restrict__ v,
                          float2* __restrict__ a)
{
  const int bz = blockIdx.y;                 // B*CO*H
  const int tile = blockIdx.x;               // (W/16)*(D/16)
  const int tm = (tile / (D_ / 16)) * 16;    // q tile
  const int tn = (tile % (D_ / 16)) * 16;    // dd tile
  const int bc = bz / H_, hh = bz % H_;
  const float2* sbase = s + (size_t)bz * W_ * W_;
  const float2* vbase = v + ((size_t)bc * F_ + (size_t)hh * D_) * W_;
  v8f sr, si;
  // A: w[m=q,k=kpos] stride(m)=W, stride(k)=1 ; B: V[k=kpos,n=dd] at dd*W+kpos
  cgemm_tile(sbase + (size_t)tm * W_, W_, 1, vbase + (size_t)tn * W_, 1, W_, W_, sr, si);
  const int lane = threadIdx.x & 31, half = lane >> 4, l16 = lane & 15;
  float2* abase = a + ((size_t)bc * F_ + (size_t)hh * D_) * W_;
#pragma unroll
  for (int j = 0; j < 8; ++j) {
    int m = tm + j + 8 * half;   // q
    int n = tn + l16;            // dd
    abase[(size_t)n * W_ + m] = make_float2(sr[j], si[j]);
  }
}

// ---------------------------------------------------------------------------
// 7) Per-channel projection: out[bc][m,w] = sum_k wgt[c][m,k]*act[bc][k,w]
// ---------------------------------------------------------------------------
__global__ void chanproj_kernel(const float2* __restrict__ wgt, const float2* __restrict__ act,
                                float2* __restrict__ out, int M, int K)
{
  const int bc = blockIdx.z;
  const int c = bc % CO_;
  const int tm = blockIdx.y * 16, tn = blockIdx.x * 16;
  v8f sr, si;
  cgemm_tile(wgt + (size_t)c * M * K + (size_t)tm * K, K, 1,
             act + (size_t)bc * K * W_ + tn, W_, 1, K, sr, si);
  const int lane = threadIdx.x & 31, half = lane >> 4, l16 = lane & 15;
  float2* O = out + (size_t)bc * M * W_;
#pragma unroll
  for (int j = 0; j < 8; ++j) {
    int m = tm + j + 8 * half, n = tn + l16;
    O[(size_t)m * W_ + n] = make_float2(sr[j], si[j]);
  }
}

// ---------------------------------------------------------------------------
// 8) h += za + zb
// ---------------------------------------------------------------------------
__global__ void add3_kernel(float2* __restrict__ h, const float2* __restrict__ za,
                            const float2* __restrict__ zb)
{
  size_t i = (size_t)blockIdx.x * blockDim.x + threadIdx.x;
  if (i >= (size_t)B_ * CO_ * F_ * W_) return;
  float2 a = h[i], x = za[i], y = zb[i];
  h[i] = make_float2(a.x + x.x + y.x, a.y + x.y + y.y);
}

// ---------------------------------------------------------------------------
// 9) FFN-1: hid = sqrelu(c1w @ z3), complex WMMA + squared-ReLU epilogue.
// ---------------------------------------------------------------------------
__global__ void ffn1_kernel(const float2* __restrict__ c1w, const float2* __restrict__ z,
                            float2* __restrict__ hid)
{
  const int bc = blockIdx.z;
  const int c = bc % CO_;
  const int tm = blockIdx.y * 16, tn = blockIdx.x * 16;
  v8f sr, si;
  cgemm_tile(c1w + (size_t)c * E_ * F_ + (size_t)tm * F_, F_, 1,
             z + (size_t)bc * F_ * W_ + tn, W_, 1, F_, sr, si);
  const int lane = threadIdx.x & 31, half = lane >> 4, l16 = lane & 15;
  float2* O = hid + (size_t)bc * E_ * W_;
#pragma unroll
  for (int j = 0; j < 8; ++j) {
    int m = tm + j + 8 * half, n = tn + l16;
    float r = fmaxf(sr[j], 0.f), i2 = fmaxf(si[j], 0.f);
    O[(size_t)m * W_ + n] = make_float2(r * r, i2 * i2);
  }
}

// ---------------------------------------------------------------------------
// 10) FFN-2 fused with residual, writes final concat region of d_out.
// ---------------------------------------------------------------------------
__global__ void ffn2_kernel(const float2* __restrict__ c2w, const float2* __restrict__ hid,
                            const float2* __restrict__ hres, float2* __restrict__ outc)
{
  const int bc = blockIdx.z;
  const int b = bc / CO_, c = bc % CO_;
  const int tm = blockIdx.y * 16, tn = blockIdx.x * 16;
  v8f sr, si;
  cgemm_tile(c2w + (size_t)c * F_ * E_ + (size_t)tm * E_, E_, 1,
             hid + (size_t)bc * E_ * W_ + tn, W_, 1, E_, sr, si);
  const int lane = threadIdx.x & 31, half = lane >> 4, l16 = lane & 15;
  const float2* Hr = hres + (size_t)bc * F_ * W_;
  float2* O = outc + ((size_t)b * (CI_ + CO_) + CI_ + c) * F_ * W_;
#pragma unroll
  for (int j = 0; j < 8; ++j) {
    int m = tm + j + 8 * half, n = tn + l16;
    float2 hv = Hr[(size_t)m * W_ + n];
    O[(size_t)m * W_ + n] = make_float2(hv.x + sr[j], hv.y + si[j]);
  }
}

// ---------------------------------------------------------------------------
extern "C" void kernel_launch(void* const* d_in, const int* in_sizes, int n_in,
                              void* d_out, int out_size, void* d_ws, size_t ws_size,
                              hipStream_t stream)
{
  (void)in_sizes; (void)n_in; (void)out_size; (void)ws_size;
  const float2* x       = (const float2*)d_in[0];
  const float2* skip    = (const float2*)d_in[1];
  const float2* embed_w = (const float2*)d_in[2];
  const float2* embed_b = (const float2*)d_in[3];
  const float2* n1w = (const float2*)d_in[4];
  const float2* n1b = (const float2*)d_in[5];
  const float2* n2w = (const float2*)d_in[6];
  const float2* n2b = (const float2*)d_in[7];
  const float2* n3w = (const float2*)d_in[8];
  const float2* n3b = (const float2*)d_in[9];
  const float2* a1qw = (const float2*)d_in[10];
  const float2* a1qb = (const float2*)d_in[11];
  const float2* a1kw = (const float2*)d_in[12];
  const float2* a1kb = (const float2*)d_in[13];
  const float2* a1vw = (const float2*)d_in[14];
  const float2* a1vb = (const float2*)d_in[15];
  const float2* a1ow = (const float2*)d_in[16];
  const float2* a2qw = (const float2*)d_in[17];
  const float2* a2qb = (const float2*)d_in[18];
  const float2* a2kw = (const float2*)d_in[19];
  const float2* a2kb = (const float2*)d_in[20];
  const float2* a2vw = (const float2*)d_in[21];
  const float2* a2vb = (const float2*)d_in[22];
  const float2* a2ow = (const float2*)d_in[23];
  const float2* c1w  = (const float2*)d_in[24];
  const float2* c2w  = (const float2*)d_in[25];

  // Output: concat(B, CI+CO, F, W) complex64, then qkm, qkp (f32, B*CO*H*W*W each)
  float2* outc = (float2*)d_out;
  float*  qkm  = (float*)d_out + 2 * (size_t)B_ * (CI_ + CO_) * F_ * W_;
  float*  qkp  = qkm + (size_t)B_ * CO_ * H_ * W_ * W_;

  // Workspace layout (complex elements)
  const size_t N1 = (size_t)B_ * CO_ * F_ * W_;              // 4.19M complex
  const size_t NS = (size_t)B_ * CO_ * H_ * W_ * W_;         // 8.39M complex
  float2* ws  = (float2*)d_ws;
  float2* h   = ws + 0 * N1;
  float2* z   = ws + 1 * N1;
  float2* q   = ws + 2 * N1;
  float2* k   = ws + 3 * N1;
  float2* v   = ws + 4 * N1;
  float2* a   = ws + 5 * N1;   // attn output; also reused as z2 between attns
  float2* za  = ws + 6 * N1;
  float2* zb  = ws + 7 * N1;
  float2* s   = ws + 8 * N1;
  float2* hid = s + NS;

  const dim3 b256(256);
  const dim3 cgrid(F_ / 16, W_ / 16, B_ * CO_);                 // conv tiles
  const dim3 sgrid((W_ / 16) * (W_ / 16), B_ * CO_ * H_);       // score tiles
  const dim3 agrid((W_ / 16) * (D_ / 16), B_ * CO_ * H_);       // attn.V tiles
  const dim3 pgrid(W_ / 16, F_ / 16, B_ * CO_);                 // FxF proj tiles
  const dim3 f1grid(W_ / 16, E_ / 16, B_ * CO_);                // FFN1 tiles
  const dim3 wave(32);

  // ---- h = embed(x); copy x into concat output -------------------------
  embed_kernel<<<dim3((unsigned)((B_ * F_ * W_ + 255) / 256)), b256, 0, stream>>>(
      x, embed_w, embed_b, h, outc);
  // ---- z = LN1(h) ------------------------------------------------------
  cln_kernel<<<dim3(W_ / 32, B_ * CO_), b256, 0, stream>>>(h, n1w, n1b, z);

  // ---- attention 1 (self) ---------------------------------------------
  cconv3_kernel<<<cgrid, b256, 0, stream>>>(z, a1qw, a1qb, q);
  cconv3_kernel<<<cgrid, b256, 0, stream>>>(z, a1kw, a1kb, k);
  cconv3_kernel<<<cgrid, b256, 0, stream>>>(z, a1vw, a1vb, v);
  scores_kernel<<<sgrid, wave, 0, stream>>>(q, k, s, qkm, qkp, 1);
  softmax_phase_kernel<<<dim3(B_ * CO_ * H_ * W_), b256, 0, stream>>>(s);
  av_kernel<<<agrid, wave, 0, stream>>>(s, v, a);
  chanproj_kernel<<<pgrid, wave, 0, stream>>>(a1ow, a, za, F_, F_);

  // ---- attention 2 (cross with skip) ----------------------------------
  float2* z2 = a;  // a no longer needed; reuse
  cln_kernel<<<dim3(W_ / 32, B_ * CO_), b256, 0, stream>>>(z, n2w, n2b, z2);
  cconv3_kernel<<<cgrid, b256, 0, stream>>>(z2, a2qw, a2qb, q);
  cconv3_kernel<<<cgrid, b256, 0, stream>>>(skip, a2kw, a2kb, k);
  cconv3_kernel<<<cgrid, b256, 0, stream>>>(skip, a2vw, a2vb, v);
  scores_kernel<<<sgrid, wave, 0, stream>>>(q, k, s, qkm, qkp, 0);
  softmax_phase_kernel<<<dim3(B_ * CO_ * H_ * W_), b256, 0, stream>>>(s);
  av_kernel<<<agrid, wave, 0, stream>>>(s, v, a);   // overwrites z2 (done with it)
  chanproj_kernel<<<pgrid, wave, 0, stream>>>(a2ow, a, zb, F_, F_);

  // ---- residual + FFN --------------------------------------------------
  add3_kernel<<<dim3((unsigned)((N1 + 255) / 256)), b256, 0, stream>>>(h, za, zb);
  cln_kernel<<<dim3(W_ / 32, B_ * CO_), b256, 0, stream>>>(h, n3w, n3b, z);
  ffn1_kernel<<<f1grid, wave, 0, stream>>>(c1w, z, hid);
  ffn2_kernel<<<pgrid, wave, 0, stream>>>(c2w, hid, h, outc);
}